// ComplExDecoder_51101520888487
// MI455X (gfx1250) — compile-verified
//
#include <hip/hip_runtime.h>

typedef float v2f __attribute__((ext_vector_type(2)));
typedef float v8f __attribute__((ext_vector_type(8)));

#define BB 2
#define NN 1000
#define CC 128
#define RC 50
#define NT 63            // ceil(1000/16)
#define TILES_PER_B (NT * NT)
#define TOTAL_TILES (BB * TILES_PER_B)

// One wave computes one 16x16 (s,o) tile of the complex Gram matrix with
// f32 WMMA, then fuses the broadcast against R[r,o] for all r, streaming
// 50 * 16x16 f32 results straight to HBM.
__global__ __launch_bounds__(256) void complex_decoder_kernel(
    const float* __restrict__ xr, const float* __restrict__ xi,
    const float* __restrict__ Rr, const float* __restrict__ Ri,
    float* __restrict__ out)
{
    const int lane = threadIdx.x & 31;
    const int wave = threadIdx.x >> 5;
    const int tile = blockIdx.x * 8 + wave;
    if (tile >= TOTAL_TILES) return;          // wave-uniform exit (EXEC stays all-1s for WMMA)

    const int b  = tile / TILES_PER_B;
    const int t  = tile - b * TILES_PER_B;
    const int st = t / NT;
    const int ot = t - st * NT;
    const int s0 = st * 16;
    const int o0 = ot * 16;

    const int m       = lane & 15;            // M (for A) / N (for B) row within tile
    const int halfsel = lane >> 4;            // 0: K pair {0,1}; 1: K pair {2,3}
    const int colBase = halfsel * 2;

    // Clamp edge rows: garbage contributions land only in rows/cols we never store.
    int arow = s0 + m; if (arow >= NN) arow = NN - 1;
    int brow = o0 + m; if (brow >= NN) brow = NN - 1;

    const float* aR = xr + ((size_t)b * NN + arow) * CC + colBase;
    const float* aI = xi + ((size_t)b * NN + arow) * CC + colBase;
    const float* bR = xr + ((size_t)b * NN + brow) * CC + colBase;
    const float* bI = xi + ((size_t)b * NN + brow) * CC + colBase;

    v8f gr = {};   // Gr = xr_s.xr_o + xi_s.xi_o
    v8f w  = {};   // W  = xr_s.xi_o
    v8f h  = {};   // H  = xi_s.xr_o       (Gi = W - H)

    #pragma unroll 8
    for (int k = 0; k < CC; k += 4) {
        v2f ar = *(const v2f*)(aR + k);
        v2f ai = *(const v2f*)(aI + k);
        v2f br = *(const v2f*)(bR + k);
        v2f bi = *(const v2f*)(bI + k);
        // 8-arg form: (neg_a, A, neg_b, B, c_mod, C, reuse_a, reuse_b)
        gr = __builtin_amdgcn_wmma_f32_16x16x4_f32(false, ar, false, br, (short)0, gr, false, false);
        gr = __builtin_amdgcn_wmma_f32_16x16x4_f32(false, ai, false, bi, (short)0, gr, false, false);
        w  = __builtin_amdgcn_wmma_f32_16x16x4_f32(false, ar, false, bi, (short)0, w,  false, false);
        h  = __builtin_amdgcn_wmma_f32_16x16x4_f32(false, ai, false, br, (short)0, h,  false, false);
    }

    // score = Gr*Rr - Gi*Ri = Gr*Rr + (H - W)*Ri
    v8f gd;
    #pragma unroll
    for (int j = 0; j < 8; ++j) gd[j] = h[j] - w[j];

    // C/D layout: VGPR j -> M = j + 8*halfsel, lane (L&15) -> N.
    const int oc     = o0 + m;
    const bool oval  = (oc < NN);
    const int ocl    = oval ? oc : (NN - 1);
    const int sbase  = s0 + halfsel * 8;

    for (int r = 0; r < RC; ++r) {
        const float rr = Rr[(size_t)r * NN + ocl];
        const float ri = Ri[(size_t)r * NN + ocl];
        // out[((b*N + sr)*R + r)*N + oc]
        const size_t obase = (((size_t)b * NN) * RC + (size_t)r) * NN + (size_t)oc;
        #pragma unroll
        for (int j = 0; j < 8; ++j) {
            const int sr = sbase + j;
            if (oval && sr < NN) {
                out[obase + (size_t)sr * ((size_t)RC * NN)] = fmaf(gd[j], ri, gr[j] * rr);
            }
        }
    }
}

extern "C" void kernel_launch(void* const* d_in, const int* in_sizes, int n_in,
                              void* d_out, int out_size, void* d_ws, size_t ws_size,
                              hipStream_t stream) {
    (void)in_sizes; (void)n_in; (void)out_size; (void)d_ws; (void)ws_size;
    const float* xr = (const float*)d_in[0];
    const float* xi = (const float*)d_in[1];
    const float* Rr = (const float*)d_in[2];
    const float* Ri = (const float*)d_in[3];
    float* out = (float*)d_out;

    const int blocks = (TOTAL_TILES + 7) / 8;   // 8 waves (tiles) per 256-thread block
    hipLaunchKernelGGL(complex_decoder_kernel, dim3(blocks), dim3(256), 0, stream,
                       xr, xi, Rr, Ri, out);
}